// DOGSS_60687887893254
// MI455X (gfx1250) — compile-verified
//
#include <hip/hip_runtime.h>
#include <math.h>

// ---------------------------------------------------------------------------
// DOGSS crystal-graph conv for MI455X (gfx1250, wave32, WMMA).
// Memory-bound workload (~180 GFLOP vs several GB traffic @23.3TB/s) -> use
// the full-precision fp32 WMMA path (V_WMMA_F32_16X16X4_F32), fuse the z-row
// gather + BN-statistics accumulation into the GEMM kernels, and load all
// A/B fragments as aligned 8-byte pairs (pair-packed weights, stride-42
// padded edge rows) so every inner-loop fetch is a single global_load_b64.
// ---------------------------------------------------------------------------

typedef __attribute__((ext_vector_type(2))) float v2f;
typedef __attribute__((ext_vector_type(8))) float v8f;

#define N_ATOMS   50000
#define MNBR      12
#define OFDIM     92
#define FDIM      64
#define EDIM      41
#define EPAD      42            // edge row stride padded for 8B-aligned pairs
#define ZDIM      169           // 2*F + E
#define KPAD      172           // ZDIM padded to multiple of 4
#define CONV_CTOT 210           // 128 node-branch + 82 edge-branch columns
#define CONV_CPAD 224           // padded to 14 tiles of 16
#define FIN_CTOT  256           // 128 dist + 128 const columns
#define NROWS     (N_ATOMS * MNBR)   // 600000 rows; 37500 exact 16-row tiles
#define BN_EPS    1e-5f

__device__ __forceinline__ float lrelu(float x) { return x > 0.f ? x : 0.01f * x; }
__device__ __forceinline__ float sigm(float x)  { return 1.f / (1.f + expf(-x)); }
__device__ __forceinline__ float softpl(float x){ return x > 20.f ? x : log1pf(expf(x)); }

__device__ __forceinline__ v8f v8zero() {
    v8f z;
#pragma unroll
    for (int i = 0; i < 8; ++i) z[i] = 0.f;
    return z;
}

// fp32 WMMA: D(16x16) = A(16x4) * B(4x16) + C, full wave32 cooperative op.
__device__ __forceinline__ v8f wmma_f32(v2f a, v2f b, v8f c) {
    return __builtin_amdgcn_wmma_f32_16x16x4_f32(
        /*neg_a=*/false, a, /*neg_b=*/false, b,
        /*c_mod=*/(short)0, c, /*reuse_a=*/false, /*reuse_b=*/false);
}

// z-row K pair (k even): [ node[n] | node[eidx] | edge_fea(stride 42, pad=0) ]
// Pairs never straddle the 64/128 boundaries; (168,169) reads the zeroed pad.
__device__ __forceinline__ v2f zpair(const float* __restrict__ np,
                                     const float* __restrict__ nbp,
                                     const float* __restrict__ ep, int k) {
    if (k < FDIM)      return *(const v2f*)(np + k);
    if (k < 2 * FDIM)  return *(const v2f*)(nbp + (k - FDIM));
    if (k <= 168)      return *(const v2f*)(ep + (k - 2 * FDIM));
    v2f z; z.x = 0.f; z.y = 0.f;
    return z;
}

// ---------------------------------------------------------------------------
// Utility kernels
// ---------------------------------------------------------------------------
__global__ void zero_kernel(float* __restrict__ p, int n) {
    int i = blockIdx.x * blockDim.x + threadIdx.x;
    if (i < n) p[i] = 0.f;
}

__global__ void finalize_stats(const float* __restrict__ sum,
                               const float* __restrict__ sq,
                               float* __restrict__ mean,
                               float* __restrict__ inv,
                               int C, float invcnt) {
    int c = blockIdx.x * blockDim.x + threadIdx.x;
    if (c >= C) return;
    float m = sum[c] * invcnt;
    float v = sq[c] * invcnt - m * m;
    mean[c] = m;
    inv[c]  = rsqrtf(v + BN_EPS);
}

// copy edge features into the mutable stride-42 buffer, zeroing the pad lane
__global__ void copy_edge(const float* __restrict__ src, float* __restrict__ dst) {
    size_t i = (size_t)blockIdx.x * blockDim.x + threadIdx.x;
    if (i >= (size_t)NROWS * EPAD) return;
    int c = (int)(i % EPAD);
    size_t row = i / EPAD;
    dst[i] = (c < EDIM) ? src[row * EDIM + c] : 0.f;
}

// distance(n,m) = |pos[eidx] + nbr_offset@cells - pos[n]| with 1e-12 fudge
__global__ void dist_kernel(const float* __restrict__ nbr_off,
                            const float* __restrict__ pos,
                            const float* __restrict__ cells,
                            const int* __restrict__ eidx,
                            float* __restrict__ dist) {
    int i = blockIdx.x * blockDim.x + threadIdx.x;
    if (i >= NROWS) return;
    int n = i / MNBR;
    float o0 = nbr_off[(size_t)i * 3 + 0];
    float o1 = nbr_off[(size_t)i * 3 + 1];
    float o2 = nbr_off[(size_t)i * 3 + 2];
    const float* C = cells + (size_t)n * 9;
    int e = eidx[i];
    float acc = 1e-12f;
#pragma unroll
    for (int j = 0; j < 3; ++j) {
        float oc = o0 * C[0 * 3 + j] + o1 * C[1 * 3 + j] + o2 * C[2 * 3 + j];
        float df = pos[(size_t)e * 3 + j] + oc - pos[(size_t)n * 3 + j];
        acc += df * df;
    }
    dist[i] = sqrtf(acc);
}

// ---------------------------------------------------------------------------
// Weight packing into WMMA-fragment pair layout:
//   Wpf[(q*Cpad + col)*2 + j] = W(2q + j, col),  q = k/2 in [0, KPAD/2)
// so a lane's B fragment (k, k+1) at column col is one aligned v2f load.
// ---------------------------------------------------------------------------
__global__ void pack_emb(const float* __restrict__ W, float* __restrict__ Wp) {
    int i = blockIdx.x * blockDim.x + threadIdx.x;      // over (OFDIM/2)*FDIM
    if (i >= (OFDIM / 2) * FDIM) return;
    int q = i / FDIM, col = i % FDIM;
    Wp[(size_t)i * 2 + 0] = W[(size_t)(2 * q + 0) * FDIM + col];
    Wp[(size_t)i * 2 + 1] = W[(size_t)(2 * q + 1) * FDIM + col];
}

__device__ __forceinline__ float conv_w(const float* Wn, const float* We, int k, int c) {
    if (k >= ZDIM) return 0.f;
    if (c < 2 * FDIM)       return Wn[(size_t)k * (2 * FDIM) + c];
    if (c < CONV_CTOT)      return We[(size_t)k * (2 * EDIM) + (c - 2 * FDIM)];
    return 0.f;
}

__global__ void pack_conv(const float* __restrict__ Wn, const float* __restrict__ bn,
                          const float* __restrict__ We, const float* __restrict__ be,
                          float* __restrict__ Wp, float* __restrict__ bp) {
    int i = blockIdx.x * blockDim.x + threadIdx.x;      // over (KPAD/2)*CONV_CPAD
    if (i < (KPAD / 2) * CONV_CPAD) {
        int q = i / CONV_CPAD, c = i % CONV_CPAD;
        Wp[(size_t)i * 2 + 0] = conv_w(Wn, We, 2 * q + 0, c);
        Wp[(size_t)i * 2 + 1] = conv_w(Wn, We, 2 * q + 1, c);
    }
    if (i < CONV_CPAD) {
        bp[i] = (i < 2 * FDIM) ? bn[i] : (i < CONV_CTOT ? be[i - 2 * FDIM] : 0.f);
    }
}

__device__ __forceinline__ float fin_w(const float* Wd, const float* Wc, int k, int c) {
    if (k >= ZDIM) return 0.f;
    return (c < 128) ? Wd[(size_t)k * 128 + c] : Wc[(size_t)k * 128 + (c - 128)];
}

__global__ void pack_final(const float* __restrict__ Wd, const float* __restrict__ bd,
                           const float* __restrict__ Wc, const float* __restrict__ bc,
                           float* __restrict__ Wp, float* __restrict__ bp) {
    int i = blockIdx.x * blockDim.x + threadIdx.x;      // over (KPAD/2)*FIN_CTOT
    if (i < (KPAD / 2) * FIN_CTOT) {
        int q = i / FIN_CTOT, c = i % FIN_CTOT;
        Wp[(size_t)i * 2 + 0] = fin_w(Wd, Wc, 2 * q + 0, c);
        Wp[(size_t)i * 2 + 1] = fin_w(Wd, Wc, 2 * q + 1, c);
    }
    if (i < FIN_CTOT) bp[i] = (i < 128) ? bd[i] : bc[i - 128];
}

// ---------------------------------------------------------------------------
// Embedding GEMM: node = node_fea(50000x92) @ W_emb(92x64) + b  via fp32 WMMA.
// 3125 row tiles x 4 col tiles; one wave per col tile; K=92 = 23 exact steps.
// ---------------------------------------------------------------------------
__global__ void gemm_embed(const float* __restrict__ nf,
                           const float* __restrict__ Wp,   // pair-packed
                           const float* __restrict__ b,
                           float* __restrict__ node) {
    const int lane = threadIdx.x & 31;
    const int wave = threadIdx.x >> 5;      // 0..3 -> col tile
    const int lrow = lane & 15;
    const int half = lane >> 4;
    const int rb   = blockIdx.x * 16;
    const int arow = rb + lrow;
    const float* ap = nf + (size_t)arow * OFDIM;
    const int col = (wave << 4) + lrow;
    const v2f* wq = (const v2f*)Wp;

    v8f acc = v8zero();
    for (int ks = 0; ks < OFDIM / 4; ++ks) {
        const int q = ks * 2 + half;
        v2f a = *(const v2f*)(ap + 2 * q);
        v2f bb = wq[(size_t)q * FDIM + col];
        acc = wmma_f32(a, bb, acc);
    }
    const float bv = b[col];
#pragma unroll
    for (int r = 0; r < 8; ++r)
        node[(size_t)(rb + r + half * 8) * FDIM + col] = acc[r] + bv;
}

// ---------------------------------------------------------------------------
// Fused gathered-z GEMM + BN-stats: raw = z @ W + bias; atomically accumulate
// per-channel sum / sum-of-squares (register partials across a persistent
// grid-stride row-tile loop; one atomic pair per lane at the end).
// z rows are built on the fly from node / node[edge_idx] / edge_fea (pairs).
// ---------------------------------------------------------------------------
__global__ void gemm_z_stats(const float* __restrict__ node,
                             const float* __restrict__ edge,   // stride EPAD
                             const int* __restrict__ eidx,
                             const float* __restrict__ Wp,     // pair-packed
                             const float* __restrict__ biasp,
                             float* __restrict__ raw,
                             float* __restrict__ sum,
                             float* __restrict__ sumsq,
                             int Ctot, int Cpad) {
    const int lane = threadIdx.x & 31;
    const int wave = threadIdx.x >> 5;      // 4 waves / block
    const int lrow = lane & 15;
    const int half = lane >> 4;
    const int CT   = Cpad >> 4;
    const int numTiles = NROWS / 16;        // 37500 exact
    const v2f* wq = (const v2f*)Wp;

    int cts[4]; int nct = 0;
    for (int ct = wave; ct < CT; ct += 4) cts[nct++] = ct;

    float s[4]  = {0.f, 0.f, 0.f, 0.f};
    float sq[4] = {0.f, 0.f, 0.f, 0.f};

    for (int tile = blockIdx.x; tile < numTiles; tile += gridDim.x) {
        const int rb   = tile * 16;
        const int arow = rb + lrow;
        const int n    = arow / MNBR;
        const int ei   = eidx[arow];
        const float* np  = node + (size_t)n  * FDIM;
        const float* nbp = node + (size_t)ei * FDIM;
        const float* ep  = edge + (size_t)arow * EPAD;
        __builtin_prefetch(ep, 0, 1);       // global_prefetch for edge row

        v8f acc[4];
        for (int i = 0; i < nct; ++i) acc[i] = v8zero();

        for (int ks = 0; ks < KPAD / 4; ++ks) {
            const int q = ks * 2 + half;
            v2f a = zpair(np, nbp, ep, 2 * q);
            const v2f* wrow = wq + (size_t)q * Cpad;
            for (int i = 0; i < nct; ++i) {
                v2f b = wrow[(cts[i] << 4) + lrow];
                acc[i] = wmma_f32(a, b, acc[i]);
            }
        }
        for (int i = 0; i < nct; ++i) {
            const int col = (cts[i] << 4) + lrow;
            if (col < Ctot) {
                const float bv = biasp[col];
#pragma unroll
                for (int r = 0; r < 8; ++r) {
                    float v = acc[i][r] + bv;
                    raw[(size_t)(rb + r + half * 8) * Ctot + col] = v;
                    s[i]  += v;
                    sq[i] += v * v;
                }
            }
        }
    }
    for (int i = 0; i < nct; ++i) {
        const int col = (cts[i] << 4) + lrow;
        if (col < Ctot) {
            atomicAdd(&sum[col],   s[i]);
            atomicAdd(&sumsq[col], sq[i]);
        }
    }
}

// ---------------------------------------------------------------------------
// Node branch: normalize raw[:,:128], gate sig(nf)*lrelu(nc), sum over M,
// accumulate stats of the (N,64) aggregate for the second BN (axis 0).
// ---------------------------------------------------------------------------
__global__ void node_gate_aggr(const float* __restrict__ raw,
                               const float* __restrict__ meanA,
                               const float* __restrict__ invA,
                               float* __restrict__ aggr,
                               float* __restrict__ bsum,
                               float* __restrict__ bsq) {
    __shared__ float ssum[64], ssq[64];
    const int t  = threadIdx.x;
    const int j  = t & 63;
    const int nl = t >> 6;                  // 4 atoms per block-iteration
    if (t < 64) { ssum[t] = 0.f; ssq[t] = 0.f; }
    __syncthreads();

    const float mf = meanA[j],      vf = invA[j];
    const float mc = meanA[64 + j], vc = invA[64 + j];
    float ls = 0.f, lss = 0.f;
    for (int n = blockIdx.x * 4 + nl; n < N_ATOMS; n += gridDim.x * 4) {
        float acc = 0.f;
        size_t base = (size_t)n * MNBR * CONV_CTOT;
        for (int m = 0; m < MNBR; ++m) {
            float f = (raw[base + (size_t)m * CONV_CTOT + j]      - mf) * vf;
            float c = (raw[base + (size_t)m * CONV_CTOT + 64 + j] - mc) * vc;
            acc += sigm(f) * lrelu(c);
        }
        aggr[(size_t)n * 64 + j] = acc;
        ls  += acc;
        lss += acc * acc;
    }
    atomicAdd(&ssum[j], ls);
    atomicAdd(&ssq[j], lss);
    __syncthreads();
    if (t < 64) {
        atomicAdd(&bsum[t], ssum[t]);
        atomicAdd(&bsq[t],  ssq[t]);
    }
}

__global__ void node_update(float* __restrict__ node,
                            const float* __restrict__ aggr,
                            const float* __restrict__ mB,
                            const float* __restrict__ iB) {
    int i = blockIdx.x * blockDim.x + threadIdx.x;
    if (i >= N_ATOMS * FDIM) return;
    int c = i & 63;
    float v = (aggr[i] - mB[c]) * iB[c];
    node[i] = lrelu(node[i] + v);
}

// ---------------------------------------------------------------------------
// Edge branch: normalize raw[:,128:210], gate, store gated (N,M,41) and its
// channel stats for the second BN over (0,1).
// ---------------------------------------------------------------------------
__global__ void edge_gate(const float* __restrict__ raw,
                          const float* __restrict__ meanA,
                          const float* __restrict__ invA,
                          float* __restrict__ gated,
                          float* __restrict__ csum,
                          float* __restrict__ csq) {
    __shared__ float ssum[EDIM], ssq[EDIM];
    const int t  = threadIdx.x;
    const int c  = t & 63;
    const int rl = t >> 6;                  // 4 rows per block-iteration
    const bool act = (c < EDIM);
    if (t < EDIM) { ssum[t] = 0.f; ssq[t] = 0.f; }
    __syncthreads();

    float ls = 0.f, lss = 0.f;
    if (act) {
        const float mf = meanA[128 + c], vf = invA[128 + c];
        const float mc = meanA[169 + c], vc = invA[169 + c];
        for (int row = blockIdx.x * 4 + rl; row < NROWS; row += gridDim.x * 4) {
            size_t rb = (size_t)row * CONV_CTOT;
            float f = (raw[rb + 128 + c] - mf) * vf;
            float e = (raw[rb + 169 + c] - mc) * vc;
            float g = sigm(f) * lrelu(e);
            gated[(size_t)row * EDIM + c] = g;
            ls  += g;
            lss += g * g;
        }
        atomicAdd(&ssum[c], ls);
        atomicAdd(&ssq[c], lss);
    }
    __syncthreads();
    if (t < EDIM) {
        atomicAdd(&csum[t], ssum[t]);
        atomicAdd(&csq[t],  ssq[t]);
    }
}

__global__ void edge_update(float* __restrict__ edge,          // stride EPAD
                            const float* __restrict__ gated,   // stride EDIM
                            const float* __restrict__ mC,
                            const float* __restrict__ iC) {
    size_t i = (size_t)blockIdx.x * blockDim.x + threadIdx.x;
    if (i >= (size_t)NROWS * EDIM) return;
    int c = (int)(i % EDIM);
    size_t row = i / EDIM;
    float v = (gated[i] - mC[c]) * iC[c];
    size_t e = row * EPAD + c;
    edge[e] = lrelu(edge[e] + v);
}

// ---------------------------------------------------------------------------
// Final epilogue: softplus(BN+dist) mean -> out[:,0]; softplus(BN) mean / N.
// ---------------------------------------------------------------------------
__global__ void final_out(const float* __restrict__ raw,
                          const float* __restrict__ mean,
                          const float* __restrict__ inv,
                          const float* __restrict__ dist,
                          float* __restrict__ out) {
    int row = blockIdx.x * blockDim.x + threadIdx.x;
    if (row >= NROWS) return;
    float d = dist[row];
    size_t b = (size_t)row * FIN_CTOT;
    float s1 = 0.f, s2 = 0.f;
    for (int c = 0; c < 128; ++c)
        s1 += softpl((raw[b + c] - mean[c]) * inv[c] + d);
    for (int c = 128; c < 256; ++c)
        s2 += softpl((raw[b + c] - mean[c]) * inv[c]);
    out[(size_t)row * 2 + 0] = s1 * (1.f / 128.f);
    out[(size_t)row * 2 + 1] = s2 * (1.f / 128.f) / (float)N_ATOMS;
}

// ---------------------------------------------------------------------------
// Host orchestration
// ---------------------------------------------------------------------------
extern "C" void kernel_launch(void* const* d_in, const int* in_sizes, int n_in,
                              void* d_out, int out_size, void* d_ws, size_t ws_size,
                              hipStream_t stream) {
    (void)in_sizes; (void)n_in; (void)out_size; (void)ws_size;

    const float* node_fea   = (const float*)d_in[0];
    const float* edge_in    = (const float*)d_in[1];
    const float* nbr_offset = (const float*)d_in[2];
    const float* atom_pos   = (const float*)d_in[3];
    const float* cells      = (const float*)d_in[4];
    const int*   edge_idx   = (const int*)d_in[5];
    const float* W_emb      = (const float*)d_in[6];
    const float* b_emb      = (const float*)d_in[7];
    const float* W_pre_node = (const float*)d_in[8];   // (3,169,128)
    const float* b_pre_node = (const float*)d_in[9];   // (3,128)
    const float* W_pre_edge = (const float*)d_in[10];  // (3,169,82)
    const float* b_pre_edge = (const float*)d_in[11];  // (3,82)
    const float* W_dist     = (const float*)d_in[12];
    const float* b_dist     = (const float*)d_in[13];
    const float* W_const    = (const float*)d_in[14];
    const float* b_const    = (const float*)d_in[15];
    float* out = (float*)d_out;

    // Workspace layout (floats); all offsets stay 8-byte aligned.
    float* ws = (float*)d_ws;
    size_t o = 0;
    float* dist  = ws + o; o += NROWS;                        // 0.6M
    float* node  = ws + o; o += (size_t)N_ATOMS * FDIM;       // 3.2M
    float* edge  = ws + o; o += (size_t)NROWS * EPAD;         // 25.2M (stride 42)
    float* aggr  = ws + o; o += (size_t)N_ATOMS * FDIM;       // 3.2M
    float* gated = ws + o; o += (size_t)NROWS * EDIM;         // 24.6M
    float* Wp    = ws + o; o += (size_t)KPAD * FIN_CTOT;      // 44032 (pair layout)
    float* bp    = ws + o; o += FIN_CTOT;
    float* sumA  = ws + o; o += 256;   // GEMM-output channel stats
    float* sqA   = ws + o; o += 256;
    float* meanA = ws + o; o += 256;
    float* invA  = ws + o; o += 256;
    float* sumB  = ws + o; o += 64;    // aggregate-node stats
    float* sqB   = ws + o; o += 64;
    float* meanB = ws + o; o += 64;
    float* invB  = ws + o; o += 64;
    float* sumC  = ws + o; o += 64;    // gated-edge stats
    float* sqC   = ws + o; o += 64;
    float* meanC = ws + o; o += 64;
    float* invC  = ws + o; o += 64;
    float* raw   = ws + o; o += (size_t)NROWS * FIN_CTOT;     // 153.6M

    // mutable edge features: stride-42 copy with zeroed pad lane
    copy_edge<<<(int)(((size_t)NROWS * EPAD + 255) / 256), 256, 0, stream>>>(
        edge_in, edge);

    dist_kernel<<<(NROWS + 255) / 256, 256, 0, stream>>>(
        nbr_offset, atom_pos, cells, edge_idx, dist);

    // node = node_fea @ W_emb + b_emb  (WMMA, pair-packed weights staged in Wp)
    pack_emb<<<((OFDIM / 2) * FDIM + 255) / 256, 256, 0, stream>>>(W_emb, Wp);
    gemm_embed<<<N_ATOMS / 16, 128, 0, stream>>>(node_fea, Wp, b_emb, node);

    for (int l = 0; l < 3; ++l) {
        pack_conv<<<((KPAD / 2) * CONV_CPAD + 255) / 256, 256, 0, stream>>>(
            W_pre_node + (size_t)l * ZDIM * 2 * FDIM, b_pre_node + (size_t)l * 2 * FDIM,
            W_pre_edge + (size_t)l * ZDIM * 2 * EDIM, b_pre_edge + (size_t)l * 2 * EDIM,
            Wp, bp);

        zero_kernel<<<2, 256, 0, stream>>>(sumA, 512);        // sumA + sqA
        gemm_z_stats<<<2048, 128, 0, stream>>>(
            node, edge, edge_idx, Wp, bp, raw, sumA, sqA, CONV_CTOT, CONV_CPAD);
        finalize_stats<<<1, 256, 0, stream>>>(
            sumA, sqA, meanA, invA, CONV_CTOT, 1.f / (float)NROWS);

        zero_kernel<<<1, 128, 0, stream>>>(sumB, 128);        // sumB + sqB
        node_gate_aggr<<<512, 256, 0, stream>>>(raw, meanA, invA, aggr, sumB, sqB);
        finalize_stats<<<1, 64, 0, stream>>>(
            sumB, sqB, meanB, invB, FDIM, 1.f / (float)N_ATOMS);
        node_update<<<(N_ATOMS * FDIM + 255) / 256, 256, 0, stream>>>(
            node, aggr, meanB, invB);

        zero_kernel<<<1, 128, 0, stream>>>(sumC, 128);        // sumC + sqC
        edge_gate<<<2048, 256, 0, stream>>>(raw, meanA, invA, gated, sumC, sqC);
        finalize_stats<<<1, 64, 0, stream>>>(
            sumC, sqC, meanC, invC, EDIM, 1.f / (float)NROWS);
        edge_update<<<(int)(((size_t)NROWS * EDIM + 255) / 256), 256, 0, stream>>>(
            edge, gated, meanC, invC);
    }

    // Final heads: bond_fea @ [W_dist | W_const]
    pack_final<<<((KPAD / 2) * FIN_CTOT + 255) / 256, 256, 0, stream>>>(
        W_dist, b_dist, W_const, b_const, Wp, bp);
    zero_kernel<<<2, 256, 0, stream>>>(sumA, 512);
    gemm_z_stats<<<2048, 128, 0, stream>>>(
        node, edge, edge_idx, Wp, bp, raw, sumA, sqA, FIN_CTOT, FIN_CTOT);
    finalize_stats<<<1, 256, 0, stream>>>(
        sumA, sqA, meanA, invA, FIN_CTOT, 1.f / (float)NROWS);

    final_out<<<(NROWS + 255) / 256, 256, 0, stream>>>(raw, meanA, invA, dist, out);
}